// ManyBodyAttn_52295521796554
// MI455X (gfx1250) — compile-verified
//
#include <hip/hip_runtime.h>
#include <hip/hip_bf16.h>

#define NSZ 160
#define NN (NSZ * NSZ)      // 25600
#define HID 128
#define NHEADS 8
#define HD 16               // head dim
#define LN_EPS 0.001f

typedef __attribute__((ext_vector_type(2))) float v2f;
typedef __attribute__((ext_vector_type(8))) float v8f;

static __device__ __forceinline__ v8f wmma_f32_16x16x4(v2f a, v2f b, v8f c) {
    return __builtin_amdgcn_wmma_f32_16x16x4_f32(
        /*neg_a=*/false, a, /*neg_b=*/false, b,
        /*c_mod=*/(short)0, c, /*reuse_a=*/false, /*reuse_b=*/false);
}

// ---------------------------------------------------------------------------
// Workspace layouts (head-major, WMMA-fragment friendly):
//   Qp[j][h][i][d]  : ((j*8+h)*160 + i)*16 + d     (pre-scaled by HD^-0.5)
//   Kp[j][h][n][d]  : ((j*8+h)*160 + n)*16 + d
//   Vp[j][h][d][k]  : ((j*8+h)*16 + d)*160 + k
//   Ep[j][h][n]     : (j*8+h)*160 + n
//   Gp[j][h][n]     : (j*8+h)*160 + n              (pre-sigmoided)
//   Wt[ct][ks][lane][2] : B-fragment-packed weights, ct*2048 + ks*64 + lane*2
// Channel unpack: c = d*8 + h  (reference reshape(n,n,HD,HEADS)).
// ---------------------------------------------------------------------------

// Prep kernel: pack [W_q | W_kv | W_eg] column tiles into WMMA B-fragment
// order so the projection GEMM loads one b64 per lane at constant offsets.
__global__ __launch_bounds__(32) void wpack_kernel(
    const float* __restrict__ Wq, const float* __restrict__ Wkv,
    const float* __restrict__ Weg, float* __restrict__ Wt)
{
    const int ct   = blockIdx.x;        // 0..24
    const int lane = threadIdx.x;       // 0..31
    const int hs   = lane / 16;
    const int lm   = lane % 16;

    const float* W; int wcols, n0;
    if (ct < 8)       { W = Wq;  wcols = HID;        n0 = ct * 16; }
    else if (ct < 24) { W = Wkv; wcols = 2 * HID;    n0 = (ct - 8) * 16; }
    else              { W = Weg; wcols = 2 * NHEADS; n0 = 0; }

    float* o = Wt + (size_t)ct * 2048 + lane * 2;
    for (int ks = 0; ks < 32; ++ks) {
        const int kk = ks * 4 + 2 * hs;
        o[ks * 64 + 0] = W[(size_t)(kk)     * wcols + n0 + lm];
        o[ks * 64 + 1] = W[(size_t)(kk + 1) * wcols + n0 + lm];
    }
}

// One 16x16 output tile: A from LDS (ds_load b64 pairs), B from packed Wt
// (global_load_b64, constant offsets), 32 chained K=4 WMMAs.
static __device__ __forceinline__ v8f gemm_tile(
    const float* __restrict__ sA, const float* __restrict__ wt,
    int lm, int hs)
{
    v8f acc = {0.f, 0.f, 0.f, 0.f, 0.f, 0.f, 0.f, 0.f};
    const float* pa = sA + lm * HID + 2 * hs;
    #pragma unroll
    for (int ks = 0; ks < 32; ++ks) {
        v2f a, b;
        a.x = pa[ks * 4];
        a.y = pa[ks * 4 + 1];
        b = *reinterpret_cast<const v2f*>(wt + ks * 64);
        acc = wmma_f32_16x16x4(a, b, acc);
    }
    return acc;
}

// ---------------------------------------------------------------------------
// Kernel 1: fused projections. Block = 256 (8 waves), one 16-row tile.
// Wave w statically owns: Q col-tile w, K col-tile w, V col-tile w;
// wave 0 additionally the 16-col EG tile.
// ---------------------------------------------------------------------------
__global__ __launch_bounds__(256) void proj_kernel(
    const float* __restrict__ feat1, const float* __restrict__ feat2,
    const float* __restrict__ Wt,
    const float* __restrict__ bq, const float* __restrict__ bkv,
    const float* __restrict__ beg,
    float* __restrict__ Qp, float* __restrict__ Kp, float* __restrict__ Vp,
    float* __restrict__ Ep, float* __restrict__ Gp)
{
    __shared__ float sA1[16][HID];
    __shared__ float sA2[16][HID];

    const int row0 = blockIdx.x * 16;   // flat feat row tile
    const int tid  = threadIdx.x;

    // b128 cooperative staging of both 16x128 tiles
    for (int idx = tid; idx < 16 * (HID / 4); idx += 256) {
        const int r = idx / (HID / 4), c4 = idx % (HID / 4);
        reinterpret_cast<float4*>(&sA1[r][0])[c4] =
            reinterpret_cast<const float4*>(&feat1[(size_t)(row0 + r) * HID])[c4];
        reinterpret_cast<float4*>(&sA2[r][0])[c4] =
            reinterpret_cast<const float4*>(&feat2[(size_t)(row0 + r) * HID])[c4];
    }
    __syncthreads();

    const int wave = tid / 32;
    const int lane = tid % 32;
    const int hs   = lane / 16;
    const int lm   = lane % 16;

    // feat rows are flat (a,b): row = a*160 + b; 16 | 160 -> a const per tile.
    const int aIdx = row0 / NSZ;
    const int b0   = row0 % NSZ;
    const int bBase = b0 + 8 * hs;       // b coordinate for accum row r is bBase + r

    const int col = wave * 16 + lm;      // 0..127
    const int d   = col / 8;
    const int h   = col % 8;

    // ---- Q tile (from feat1): out (i=aIdx, j=bBase+r), channel col
    {
        v8f acc = gemm_tile(&sA1[0][0], Wt + (size_t)wave * 2048 + lane * 2, lm, hs);
        const float bv = bq[col];
        float* oq = Qp + (((size_t)bBase * NHEADS + h) * NSZ + aIdx) * HD + d;
        #pragma unroll
        for (int r = 0; r < 8; ++r)
            oq[(size_t)r * NHEADS * NSZ * HD] = (acc[r] + bv) * 0.25f;
    }

    // ---- K tile (from feat2): out (j=aIdx, k=bBase+r)
    {
        v8f acc = gemm_tile(&sA2[0][0], Wt + (size_t)(8 + wave) * 2048 + lane * 2, lm, hs);
        const float bv = bkv[col];
        float* ok = Kp + (((size_t)aIdx * NHEADS + h) * NSZ + bBase) * HD + d;
        #pragma unroll
        for (int r = 0; r < 8; ++r)
            ok[r * HD] = acc[r] + bv;
    }

    // ---- V tile (from feat2)
    {
        v8f acc = gemm_tile(&sA2[0][0], Wt + (size_t)(16 + wave) * 2048 + lane * 2, lm, hs);
        const float bv = bkv[HID + col];
        float* ov = Vp + (((size_t)aIdx * NHEADS + h) * HD + d) * NSZ + bBase;
        #pragma unroll
        for (int r = 0; r < 8; ++r)
            ov[r] = acc[r] + bv;
    }

    // ---- EG tile (wave 0 only; wave-uniform so EXEC is full for the WMMAs)
    if (wave == 0) {
        v8f acc = gemm_tile(&sA2[0][0], Wt + (size_t)24 * 2048 + lane * 2, lm, hs);
        const float bv = beg[lm];
        if (lm < NHEADS) {
            float* oe = Ep + ((size_t)aIdx * NHEADS + lm) * NSZ + bBase;
            #pragma unroll
            for (int r = 0; r < 8; ++r) oe[r] = acc[r] + bv;
        } else {
            float* og = Gp + ((size_t)aIdx * NHEADS + (lm - NHEADS)) * NSZ + bBase;
            #pragma unroll
            for (int r = 0; r < 8; ++r)
                og[r] = 1.f / (1.f + __expf(-(acc[r] + bv)));
        }
    }
}

// ---------------------------------------------------------------------------
// Kernel 2: fused attention + LayerNorm.
// grid = (j, itile); block = 256 = 8 waves; wave h = head h.
// S(16x160) = Q @ K^T + E (40 WMMA) -> softmax*gate -> P transposed via LDS
// in two 80-col halves -> Va(16x16) = P @ V (40 WMMA) -> fused LayerNorm.
// ---------------------------------------------------------------------------
__global__ __launch_bounds__(256) void attn_kernel(
    const float* __restrict__ Qp, const float* __restrict__ Kp,
    const float* __restrict__ Vp, const float* __restrict__ Ep,
    const float* __restrict__ Gp,
    const float* __restrict__ ln_g, const float* __restrict__ ln_b,
    float* __restrict__ out)
{
    __shared__ float Plds[NHEADS][16][81];
    __shared__ float VaLds[16][HID];

    const int j    = blockIdx.x;
    const int i0   = blockIdx.y * 16;
    const int tid  = threadIdx.x;
    const int h    = tid / 32;
    const int lane = tid % 32;
    const int hs   = lane / 16;
    const int lm   = lane % 16;

    const size_t jh = (size_t)j * NHEADS + h;

    // Per-lane base pointers; all loop offsets below are compile-time consts.
    const float* pQ = Qp + (jh * NSZ + (i0 + lm)) * HD + 2 * hs;
    const float* pK = Kp + (jh * NSZ + lm) * HD + 2 * hs;
    const float* pV = Vp + (jh * HD + lm) * NSZ + 2 * hs;
    const float* pE = Ep + jh * NSZ + lm;
    const float* pG = Gp + jh * NSZ + lm;

    // Prefetch this head's V slice (16*160 floats, contiguous) while we do S.
    {
        const float* vslice = Vp + jh * HD * NSZ;
        for (int p = lane; p < 80; p += 32)
            __builtin_prefetch(vslice + p * 32, 0, 1);
    }

    // ---- A fragments of Q tile (b64 loads: d pair contiguous)
    v2f afr[4];
    #pragma unroll
    for (int ks = 0; ks < 4; ++ks)
        afr[ks] = *reinterpret_cast<const v2f*>(pQ + ks * 4);

    // ---- S = Q @ K^T + E
    v8f acc[10];
    #pragma unroll
    for (int ct = 0; ct < 10; ++ct) {
        v8f c = {0.f, 0.f, 0.f, 0.f, 0.f, 0.f, 0.f, 0.f};
        #pragma unroll
        for (int ks = 0; ks < 4; ++ks) {
            v2f b = *reinterpret_cast<const v2f*>(pK + ct * 16 * HD + ks * 4);
            c = wmma_f32_16x16x4(afr[ks], b, c);
        }
        const float e = pE[ct * 16];
        #pragma unroll
        for (int r = 0; r < 8; ++r) c[r] += e;
        acc[ct] = c;
    }

    // ---- row softmax over 160 keys, then * sigmoid gate
    float gate[10];
    #pragma unroll
    for (int ct = 0; ct < 10; ++ct) gate[ct] = pG[ct * 16];

    #pragma unroll
    for (int r = 0; r < 8; ++r) {
        float mx = -1e30f;
        #pragma unroll
        for (int ct = 0; ct < 10; ++ct) mx = fmaxf(mx, acc[ct][r]);
        #pragma unroll
        for (int msk = 1; msk < 16; msk <<= 1) mx = fmaxf(mx, __shfl_xor(mx, msk, 32));
        float sm = 0.f;
        #pragma unroll
        for (int ct = 0; ct < 10; ++ct) {
            float e = __expf(acc[ct][r] - mx);
            acc[ct][r] = e;
            sm += e;
        }
        #pragma unroll
        for (int msk = 1; msk < 16; msk <<= 1) sm += __shfl_xor(sm, msk, 32);
        const float inv = 1.f / sm;
        #pragma unroll
        for (int ct = 0; ct < 10; ++ct) acc[ct][r] = acc[ct][r] * inv * gate[ct];
    }

    // ---- Va = P @ V in two 80-column halves (P transposed through LDS)
    v8f va = {0.f, 0.f, 0.f, 0.f, 0.f, 0.f, 0.f, 0.f};
    #pragma unroll
    for (int half = 0; half < 2; ++half) {
        #pragma unroll
        for (int ct = 0; ct < 5; ++ct) {
            #pragma unroll
            for (int r = 0; r < 8; ++r)
                Plds[h][r + 8 * hs][ct * 16 + lm] = acc[half * 5 + ct][r];
        }
        __syncthreads();
        #pragma unroll
        for (int k0 = 0; k0 < 80; k0 += 4) {
            v2f a, b;
            a.x = Plds[h][lm][k0 + 2 * hs];
            a.y = Plds[h][lm][k0 + 2 * hs + 1];
            b = *reinterpret_cast<const v2f*>(pV + half * 80 + k0);
            va = wmma_f32_16x16x4(a, b, va);
        }
        __syncthreads();
    }

    // ---- scatter Va into channel-packed tile: channel = d*8 + h
    #pragma unroll
    for (int r = 0; r < 8; ++r)
        VaLds[r + 8 * hs][lm * NHEADS + h] = va[r];
    __syncthreads();

    // ---- LayerNorm: 16 threads per row, 8 channels per thread
    const int rowLN = tid / 16;
    const int cg    = tid % 16;
    float vals[8], s1 = 0.f, s2 = 0.f;
    #pragma unroll
    for (int c = 0; c < 8; ++c) {
        float v = VaLds[rowLN][cg * 8 + c];
        vals[c] = v; s1 += v; s2 += v * v;
    }
    #pragma unroll
    for (int msk = 1; msk < 16; msk <<= 1) {
        s1 += __shfl_xor(s1, msk, 32);
        s2 += __shfl_xor(s2, msk, 32);
    }
    const float mu  = s1 * (1.f / HID);
    const float var = s2 * (1.f / HID) - mu * mu;
    const float inv = rsqrtf(var + LN_EPS);

    const size_t obase = ((size_t)(i0 + rowLN) * NSZ + j) * HID + cg * 8;
    v2f* op = reinterpret_cast<v2f*>(&out[obase]);   // 32B-aligned
    #pragma unroll
    for (int c = 0; c < 8; c += 2) {
        const int ch = cg * 8 + c;
        v2f t;
        t.x = (vals[c]     - mu) * inv * ln_g[ch]     + ln_b[ch];
        t.y = (vals[c + 1] - mu) * inv * ln_g[ch + 1] + ln_b[ch + 1];
        op[c / 2] = t;
    }
}

// ---------------------------------------------------------------------------
extern "C" void kernel_launch(void* const* d_in, const int* in_sizes, int n_in,
                              void* d_out, int out_size, void* d_ws, size_t ws_size,
                              hipStream_t stream) {
    const float* feat1 = (const float*)d_in[0];
    const float* feat2 = (const float*)d_in[1];
    const float* Wq    = (const float*)d_in[2];
    const float* bq    = (const float*)d_in[3];
    const float* Wkv   = (const float*)d_in[4];
    const float* bkv   = (const float*)d_in[5];
    const float* Weg   = (const float*)d_in[6];
    const float* beg   = (const float*)d_in[7];
    const float* lng   = (const float*)d_in[8];
    const float* lnb   = (const float*)d_in[9];

    float* ws = (float*)d_ws;
    float* Qp = ws;
    float* Kp = Qp + (size_t)NN * HID;
    float* Vp = Kp + (size_t)NN * HID;
    float* Ep = Vp + (size_t)NN * HID;
    float* Gp = Ep + (size_t)NN * NHEADS;
    float* Wt = Gp + (size_t)NN * NHEADS;   // 25 * 2048 floats

    wpack_kernel<<<dim3(25), 32, 0, stream>>>(Wq, Wkv, Weg, Wt);

    proj_kernel<<<dim3(NN / 16), 256, 0, stream>>>(
        feat1, feat2, Wt, bq, bkv, beg, Qp, Kp, Vp, Ep, Gp);

    attn_kernel<<<dim3(NSZ, NSZ / 16), 256, 0, stream>>>(
        Qp, Kp, Vp, Ep, Gp, lng, lnb, (float*)d_out);
}